// WithinSubjectTripletLoss_58308476010841
// MI455X (gfx1250) — compile-verified
//
#include <hip/hip_runtime.h>
#include <stdint.h>

#define NUM_CLASSES 6
#define NUM_SUBJECTS 16
#define NGROUP (NUM_CLASSES * NUM_SUBJECTS)   // 96
#define D 256
#define MARGIN 1.0f
#define EPS 1e-6f

typedef float v2f __attribute__((ext_vector_type(2)));
typedef float v8f __attribute__((ext_vector_type(8)));

__device__ __forceinline__ unsigned int orderable_f32(float x) {
    unsigned int u = __float_as_uint(x);
    return (u & 0x80000000u) ? ~u : (u | 0x80000000u);
}

// ---------------- K0: init workspace ----------------
__global__ void k_init(int* first, int* second, int* cnt, int* sbjtot,
                       unsigned long long* best, float* A) {
    int i = blockIdx.x * blockDim.x + threadIdx.x;
    if (i < NGROUP) {
        first[i]  = 0x7fffffff;
        second[i] = 0x7fffffff;
        cnt[i]    = 0;
        best[i]   = ~0ull;
    }
    if (i < NUM_SUBJECTS) sbjtot[i] = 0;
    for (int k = i; k < NGROUP * D; k += gridDim.x * blockDim.x) A[k] = 0.0f;
}

// ---------------- K1: counts + first positive ----------------
__global__ void k_count(const int* __restrict__ labels, const int* __restrict__ sbj,
                        int* first, int* cnt, int* sbjtot, int B) {
    int i = blockIdx.x * blockDim.x + threadIdx.x;
    if (i >= B) return;
    int s = sbj[i], c = labels[i];
    int g = s * NUM_CLASSES + c;
    atomicAdd(&cnt[g], 1);
    atomicMin(&first[g], i);
    atomicAdd(&sbjtot[s], 1);
}

// ---------------- K2: second positive ----------------
__global__ void k_second(const int* __restrict__ labels, const int* __restrict__ sbj,
                         const int* __restrict__ first, int* second, int B) {
    int i = blockIdx.x * blockDim.x + threadIdx.x;
    if (i >= B) return;
    int g = sbj[i] * NUM_CLASSES + labels[i];
    if (i > first[g]) atomicMin(&second[g], i);
}

// ---------------- K3: validity + gather anchor matrix A (96 x 256) -------
__global__ void k_gather(const float* __restrict__ emb, const int* __restrict__ first,
                         const int* __restrict__ cnt, const int* __restrict__ sbjtot,
                         int* valid, float* A) {
    int g = blockIdx.x;
    int d = threadIdx.x;
    int cp = cnt[g];
    int s  = g / NUM_CLASSES;
    int negc = sbjtot[s] - cp;
    if (d == 0) valid[g] = (cp >= 2 && negc >= 1) ? 1 : 0;
    if (cp >= 1) A[g * D + d] = emb[(size_t)first[g] * D + d];
}

// ---------------- K4: hard-negative mining GEMM (WMMA f32 16x16x4) -------
// grid: (B/256 j-slabs, 6 p-tiles), 256 threads (8 waves).
// Wave w handles columns [jA..jA+15] and [jB..jB+15]; accumulates two
// 16x16 fp32 tiles of A_tile (16x256) x emb^T over K=256 in steps of 4.
__global__ __launch_bounds__(256, 2)
void k_mine(const float* __restrict__ emb, const int* __restrict__ labels,
            const int* __restrict__ sbj, const float* __restrict__ A,
            unsigned long long* __restrict__ best) {
    __shared__ float As[16 * 260];            // stride 260: bank-conflict-free
    __shared__ unsigned long long bestL[16];

    const int tid   = threadIdx.x;
    const int ptile = blockIdx.y;

    // Stage this p-tile's 16x256 anchor rows into LDS.
    for (int idx = tid; idx < 16 * D; idx += 256) {
        int r = idx >> 8;
        int d = idx & (D - 1);
        As[r * 260 + d] = A[(ptile * 16 + r) * D + d];
    }
    if (tid < 16) bestL[tid] = ~0ull;
    __syncthreads();

    const int wave = tid >> 5;
    const int lane = tid & 31;
    const int n    = lane & 15;   // column within 16-group / A row M
    const int hi   = lane >> 4;   // half-wave selector
    const int kb   = hi * 2;      // K sub-offset per ISA f32 A/B layout
    const int jA   = blockIdx.x * 256 + wave * 32;
    const int jB   = jA + 16;

    const float*  pa  = &As[n * 260 + kb];
    const float2* pbA = reinterpret_cast<const float2*>(emb + (size_t)(jA + n) * D + kb);
    const float2* pbB = reinterpret_cast<const float2*>(emb + (size_t)(jB + n) * D + kb);

    v8f acc0 = {};
    v8f acc1 = {};
    float nsA = 0.0f, nsB = 0.0f;   // fused |x|^2 accumulation

    #pragma unroll 8
    for (int kk = 0; kk < D; kk += 4) {
        v2f a;
        a.x = pa[kk];
        a.y = pa[kk + 1];
        float2 b0 = pbA[kk >> 1];
        float2 b1 = pbB[kk >> 1];
        v2f vb0; vb0.x = b0.x; vb0.y = b0.y;
        v2f vb1; vb1.x = b1.x; vb1.y = b1.y;
        nsA = fmaf(b0.x, b0.x, fmaf(b0.y, b0.y, nsA));
        nsB = fmaf(b1.x, b1.x, fmaf(b1.y, b1.y, nsB));
        acc0 = __builtin_amdgcn_wmma_f32_16x16x4_f32(false, a, false, vb0,
                                                     (short)0, acc0, false, false);
        acc1 = __builtin_amdgcn_wmma_f32_16x16x4_f32(false, a, false, vb1,
                                                     (short)0, acc1, false, false);
    }

    // Full per-column squared norms: combine the two K-halves.
    float nrmA = nsA + __shfl_xor(nsA, 16, 32);
    float nrmB = nsB + __shfl_xor(nsB, 16, 32);

    int sjA = sbj[jA + n], ljA = labels[jA + n];
    int sjB = sbj[jB + n], ljB = labels[jB + n];

    #pragma unroll
    for (int r = 0; r < 8; ++r) {
        int p  = ptile * 16 + r + hi * 8;     // C row -> group id
        int sp = p / NUM_CLASSES;
        int cp = p - sp * NUM_CLASSES;
        // score = |x|^2 - 2 a.x  (|a|^2 dropped: argmin-invariant)
        float scA = nrmA - 2.0f * acc0[r];
        float scB = nrmB - 2.0f * acc1[r];
        unsigned long long kA = (sjA == sp && ljA != cp)
            ? ((unsigned long long)orderable_f32(scA) << 32) | (unsigned)(jA + n)
            : ~0ull;
        unsigned long long kB = (sjB == sp && ljB != cp)
            ? ((unsigned long long)orderable_f32(scB) << 32) | (unsigned)(jB + n)
            : ~0ull;
        unsigned long long key = kA < kB ? kA : kB;
        #pragma unroll
        for (int m = 1; m <= 8; m <<= 1) {
            unsigned long long o = __shfl_xor(key, m, 16);
            key = (o < key) ? o : key;
        }
        if (n == 0) atomicMin(&bestL[r + hi * 8], key);
    }
    __syncthreads();
    if (tid < 16) atomicMin(&best[ptile * 16 + tid], bestL[tid]);
}

// ---------------- K5: per-group triplet term ----------------
__global__ void k_final(const float* __restrict__ emb, const int* __restrict__ first,
                        const int* __restrict__ second, const int* __restrict__ valid,
                        const unsigned long long* __restrict__ best,
                        float* per_trip, float* validf) {
    __shared__ float redA[D];
    __shared__ float redN[D];
    int p = blockIdx.x;
    int d = threadIdx.x;
    if (!valid[p]) {
        if (d == 0) { per_trip[p] = 0.0f; validf[p] = 0.0f; }
        return;
    }
    int ai = first[p];
    int pi = second[p];
    int ni = (int)(best[p] & 0xffffffffull);
    float av = emb[(size_t)ai * D + d];
    float da = av - emb[(size_t)pi * D + d] + EPS;   // eps inside the norm
    float dn = av - emb[(size_t)ni * D + d] + EPS;
    redA[d] = da * da;
    redN[d] = dn * dn;
    __syncthreads();
    for (int s = D / 2; s > 0; s >>= 1) {
        if (d < s) { redA[d] += redA[d + s]; redN[d] += redN[d + s]; }
        __syncthreads();
    }
    if (d == 0) {
        float t = sqrtf(redA[0]) - sqrtf(redN[0]) + MARGIN;
        per_trip[p] = fmaxf(t, 0.0f);
        validf[p]   = 1.0f;
    }
}

// ---------------- K6: deterministic final reduction ----------------
__global__ void k_reduce(const float* __restrict__ per_trip,
                         const float* __restrict__ validf, float* out) {
    if (blockIdx.x == 0 && threadIdx.x == 0) {
        float s = 0.0f, c = 0.0f;
        for (int i = 0; i < NGROUP; ++i) { s += per_trip[i]; c += validf[i]; }
        out[0] = (c > 0.0f) ? s / fmaxf(c, 1.0f) : 0.0f;
    }
}

extern "C" void kernel_launch(void* const* d_in, const int* in_sizes, int n_in,
                              void* d_out, int out_size, void* d_ws, size_t ws_size,
                              hipStream_t stream) {
    (void)n_in; (void)out_size; (void)ws_size;
    const float* emb    = (const float*)d_in[0];
    const int*   labels = (const int*)d_in[1];
    const int*   sbj    = (const int*)d_in[2];
    float* out = (float*)d_out;
    const int B = in_sizes[1];               // 131072

    char* ws = (char*)d_ws;
    size_t off = 0;
    auto walloc = [&](size_t bytes, size_t align) -> void* {
        off = (off + align - 1) & ~(align - 1);
        void* p = ws + off;
        off += bytes;
        return p;
    };
    int* first   = (int*)walloc(NGROUP * sizeof(int), 4);
    int* second  = (int*)walloc(NGROUP * sizeof(int), 4);
    int* cnt     = (int*)walloc(NGROUP * sizeof(int), 4);
    int* sbjtot  = (int*)walloc(NUM_SUBJECTS * sizeof(int), 4);
    int* valid   = (int*)walloc(NGROUP * sizeof(int), 4);
    unsigned long long* best =
        (unsigned long long*)walloc(NGROUP * sizeof(unsigned long long), 8);
    float* A        = (float*)walloc((size_t)NGROUP * D * sizeof(float), 16);
    float* per_trip = (float*)walloc(NGROUP * sizeof(float), 4);
    float* validf   = (float*)walloc(NGROUP * sizeof(float), 4);

    k_init<<<NGROUP, 256, 0, stream>>>(first, second, cnt, sbjtot, best, A);
    k_count<<<(B + 255) / 256, 256, 0, stream>>>(labels, sbj, first, cnt, sbjtot, B);
    k_second<<<(B + 255) / 256, 256, 0, stream>>>(labels, sbj, first, second, B);
    k_gather<<<NGROUP, D, 0, stream>>>(emb, first, cnt, sbjtot, valid, A);
    k_mine<<<dim3(B / 256, NGROUP / 16), 256, 0, stream>>>(emb, labels, sbj, A, best);
    k_final<<<NGROUP, D, 0, stream>>>(emb, first, second, valid, best, per_trip, validf);
    k_reduce<<<1, 32, 0, stream>>>(per_trip, validf, out);
}